// LSTMLayer_9165460210113
// MI455X (gfx1250) — compile-verified
//
#include <hip/hip_runtime.h>

typedef __attribute__((ext_vector_type(2))) float v2f;
typedef __attribute__((ext_vector_type(8))) float v8f;

#define SEQ   512
#define BATCH 64
#define DIN   1024
#define HID   1024

#define WMMA_F32(A, B, C) \
    __builtin_amdgcn_wmma_f32_16x16x4_f32(false, (A), false, (B), (short)0, (C), false, false)

// ---------------------------------------------------------------------------
// Init kernel: h_init = h0 + noise_h ; c = c0 + noise_c   (both [64][1024])
// ---------------------------------------------------------------------------
__global__ void lstm_init(const float* __restrict__ h0,
                          const float* __restrict__ c0,
                          const float* __restrict__ noise_h,
                          const float* __restrict__ noise_c,
                          float* __restrict__ hinit,
                          float* __restrict__ cstate) {
    int idx = blockIdx.x * blockDim.x + threadIdx.x;   // 0 .. 65535
    int j = idx & (HID - 1);
    hinit[idx]  = h0[j] + noise_h[idx];
    cstate[idx] = c0[j] + noise_c[idx];
}

// ---------------------------------------------------------------------------
// One LSTM timestep.
// grid = 64 blocks, block = 256 threads (8 wave32).
// Block owns hidden columns j0 = 16*blockIdx .. +15, all 4 gates, 64 batch rows
//   -> sixteen 16x16 output tiles.
// Wave w:  kb = w & 1  : K-half (0 -> x_t @ W_ih^T, 1 -> h @ W_hh^T)
//          q  = w >> 1 : quadrant -> 2 M-tiles x 2 gates = 4 acc tiles
// K-loop is software-pipelined with a 4-stage fragment ring so ~16
// global_load_b64 stay in flight per wave, hiding L2 latency behind WMMAs.
// ---------------------------------------------------------------------------
__launch_bounds__(256)
__global__ void lstm_step(const float* __restrict__ x_t,     // [64][1024]
                          const float* __restrict__ h_prev,  // [64][1024]
                          const float* __restrict__ W_ih,    // [4096][1024]
                          const float* __restrict__ W_hh,    // [4096][1024]
                          const float* __restrict__ b_ih,    // [4096]
                          const float* __restrict__ b_hh,    // [4096]
                          float* __restrict__ c_state,       // [64][1024] in/out
                          float* __restrict__ h_out,         // [64][1024]
                          float* __restrict__ h_final,       // or nullptr
                          float* __restrict__ c_final) {     // or nullptr
    // Union buffer, 16 KB:
    //  stage A: K-half-1 partial tiles, [q][tile][16][16]
    //  stage B: gate slab, [gate][64 rows][16 cols]
    __shared__ float lds_buf[4096];

    const int tid  = threadIdx.x;
    const int wave = tid >> 5;
    const int lane = tid & 31;
    const int kb   = wave & 1;          // K half
    const int q    = wave >> 1;         // quadrant 0..3
    const int mt   = (q & 1) * 32;      // first batch row (tiles mt, mt+16)
    const int gp   = (q >> 1) * 2;      // first gate (gates gp, gp+1)
    const int kh   = lane >> 4;         // WMMA K-pair select
    const int mn   = lane & 15;         // A: row-in-tile / B: col-in-tile
    const int j0   = blockIdx.x * 16;

    // Sources for this wave's K half
    const float* asrc = kb ? h_prev : x_t;     // [64][1024]
    const float* wsrc = kb ? W_hh   : W_ih;    // [4096][1024]

    const float* a0 = asrc + (size_t)(mt + mn) * 1024;
    const float* a1 = asrc + (size_t)(mt + 16 + mn) * 1024;
    const float* b0 = wsrc + (size_t)((gp + 0) * HID + j0 + mn) * 1024;
    const float* b1 = wsrc + (size_t)((gp + 1) * HID + j0 + mn) * 1024;

    v8f acc00 = {};   // (rows mt..mt+15   , gate gp  )
    v8f acc01 = {};   // (rows mt..mt+15   , gate gp+1)
    v8f acc10 = {};   // (rows mt+16..mt+31, gate gp  )
    v8f acc11 = {};   // (rows mt+16..mt+31, gate gp+1)

    // ---- Software-pipelined K loop: 4-stage fragment ring -----------------
    v2f sa0[4], sa1[4], sb0[4], sb1[4];
    #pragma unroll
    for (int p = 0; p < 4; ++p) {              // prime stages k = 0,4,8,12
        const int kk = 4 * p + 2 * kh;
        sa0[p] = *(const v2f*)(a0 + kk);
        sa1[p] = *(const v2f*)(a1 + kk);
        sb0[p] = *(const v2f*)(b0 + kk);
        sb1[p] = *(const v2f*)(b1 + kk);
    }

    for (int k = 0; k < 1024 - 16; k += 16) {  // 63 iters, 4 k-steps each
        #pragma unroll
        for (int p = 0; p < 4; ++p) {
            const int ko = k + 16 + 4 * p + 2 * kh;   // prefetch 4 stages ahead
            v2f na0 = *(const v2f*)(a0 + ko);
            v2f na1 = *(const v2f*)(a1 + ko);
            v2f nb0 = *(const v2f*)(b0 + ko);
            v2f nb1 = *(const v2f*)(b1 + ko);
            acc00 = WMMA_F32(sa0[p], sb0[p], acc00);
            acc01 = WMMA_F32(sa0[p], sb1[p], acc01);
            acc10 = WMMA_F32(sa1[p], sb0[p], acc10);
            acc11 = WMMA_F32(sa1[p], sb1[p], acc11);
            sa0[p] = na0; sa1[p] = na1; sb0[p] = nb0; sb1[p] = nb1;
        }
    }
    #pragma unroll
    for (int p = 0; p < 4; ++p) {              // drain k = 1008..1020
        acc00 = WMMA_F32(sa0[p], sb0[p], acc00);
        acc01 = WMMA_F32(sa0[p], sb1[p], acc01);
        acc10 = WMMA_F32(sa1[p], sb0[p], acc10);
        acc11 = WMMA_F32(sa1[p], sb1[p], acc11);
    }

    // ---- Deterministic K-half reduction through LDS -----------------------
    // C/D layout: VGPR r, lanes 0-15 -> row r, col = lane; lanes 16-31 -> row r+8.
    const int crow = kh * 8;   // row base within a 16x16 tile for this lane

    if (kb == 1) {
        float* p = lds_buf + q * 1024;         // [q][tile 0..3][256]
        #pragma unroll
        for (int r = 0; r < 8; ++r) {
            p[0 * 256 + (crow + r) * 16 + mn] = acc00[r];
            p[1 * 256 + (crow + r) * 16 + mn] = acc01[r];
            p[2 * 256 + (crow + r) * 16 + mn] = acc10[r];
            p[3 * 256 + (crow + r) * 16 + mn] = acc11[r];
        }
    }
    __syncthreads();
    if (kb == 0) {
        const float* p = lds_buf + q * 1024;
        #pragma unroll
        for (int r = 0; r < 8; ++r) {
            acc00[r] += p[0 * 256 + (crow + r) * 16 + mn];
            acc01[r] += p[1 * 256 + (crow + r) * 16 + mn];
            acc10[r] += p[2 * 256 + (crow + r) * 16 + mn];
            acc11[r] += p[3 * 256 + (crow + r) * 16 + mn];
        }
    }
    __syncthreads();   // all partial reads done before slab overwrite

    // ---- kb==0 waves publish the final gate slab: [gate][64][16] ----------
    if (kb == 0) {
        float* g0 = lds_buf + (gp + 0) * (BATCH * 16);
        float* g1 = lds_buf + (gp + 1) * (BATCH * 16);
        #pragma unroll
        for (int r = 0; r < 8; ++r) {
            g0[(mt + crow + r) * 16 + mn]      = acc00[r];
            g1[(mt + crow + r) * 16 + mn]      = acc01[r];
            g0[(mt + 16 + crow + r) * 16 + mn] = acc10[r];
            g1[(mt + 16 + crow + r) * 16 + mn] = acc11[r];
        }
    }
    __syncthreads();

    // ---- Pointwise: activations, c update, h output -----------------------
    {
        const int b  = tid >> 2;            // batch row 0..63
        const int n4 = (tid & 3) * 4;       // column group within the 16
        #pragma unroll
        for (int u = 0; u < 4; ++u) {
            const int n = n4 + u;
            const int j = j0 + n;
            const int e = b * 16 + n;       // index within a gate slab
            float gi = lds_buf[0 * 1024 + e] + b_ih[0 * HID + j] + b_hh[0 * HID + j];
            float gf = lds_buf[1 * 1024 + e] + b_ih[1 * HID + j] + b_hh[1 * HID + j];
            float gg = lds_buf[2 * 1024 + e] + b_ih[2 * HID + j] + b_hh[2 * HID + j];
            float go = lds_buf[3 * 1024 + e] + b_ih[3 * HID + j] + b_hh[3 * HID + j];

            const float i_g = 1.0f / (1.0f + __expf(-gi));
            const float f_g = 1.0f / (1.0f + __expf(-gf));
            const float g_g = 1.0f - 2.0f / (__expf(2.0f * gg) + 1.0f);   // tanh
            const float o_g = 1.0f / (1.0f + __expf(-go));

            const size_t off = (size_t)b * HID + j;
            const float cp = c_state[off];
            const float cn = f_g * cp + i_g * g_g;
            const float hn = o_g * (1.0f - 2.0f / (__expf(2.0f * cn) + 1.0f));

            c_state[off] = cn;
            h_out[off]   = hn;
            if (c_final) {               // last timestep: also emit final (h, c)
                h_final[off] = hn;
                c_final[off] = cn;
            }
        }
    }
}

// ---------------------------------------------------------------------------
// Host-side launcher (graph-capture safe: only stream-ordered launches).
// ---------------------------------------------------------------------------
extern "C" void kernel_launch(void* const* d_in, const int* in_sizes, int n_in,
                              void* d_out, int out_size, void* d_ws, size_t ws_size,
                              hipStream_t stream) {
    const float* input   = (const float*)d_in[0];  // [512][64][1024]
    const float* W_ih    = (const float*)d_in[1];  // [4096][1024]
    const float* W_hh    = (const float*)d_in[2];  // [4096][1024]
    const float* b_ih    = (const float*)d_in[3];  // [4096]
    const float* b_hh    = (const float*)d_in[4];  // [4096]
    const float* h0      = (const float*)d_in[5];  // [1024]
    const float* c0      = (const float*)d_in[6];  // [1024]
    const float* noise_h = (const float*)d_in[7];  // [64][1024]
    const float* noise_c = (const float*)d_in[8];  // [64][1024]

    float* outputs = (float*)d_out;                           // [512][64][1024]
    float* h_final = outputs + (size_t)SEQ * BATCH * HID;     // [64][1024]
    float* c_final = h_final + (size_t)BATCH * HID;           // [64][1024]

    float* hinit  = (float*)d_ws;                             // [64][1024]
    float* cstate = hinit + (size_t)BATCH * HID;              // [64][1024]

    lstm_init<<<(BATCH * HID) / 256, 256, 0, stream>>>(
        h0, c0, noise_h, noise_c, hinit, cstate);

    for (int t = 0; t < SEQ; ++t) {
        const float* x_t    = input + (size_t)t * BATCH * DIN;
        const float* h_prev = (t == 0) ? hinit
                                       : outputs + (size_t)(t - 1) * BATCH * HID;
        float* h_out = outputs + (size_t)t * BATCH * HID;
        const bool last = (t == SEQ - 1);
        lstm_step<<<HID / 16, 256, 0, stream>>>(
            x_t, h_prev, W_ih, W_hh, b_ih, b_hh,
            cstate, h_out,
            last ? h_final : nullptr,
            last ? c_final : nullptr);
    }
}